// OESNN_SEPhIA_MultiTiled2_53377853554813
// MI455X (gfx1250) — compile-verified
//
#include <hip/hip_runtime.h>
#include <hip/hip_bf16.h>

typedef __attribute__((ext_vector_type(2))) float v2f;
typedef __attribute__((ext_vector_type(8))) float v8f;

#define NT 32           // timesteps
#define NB 8192         // batch
#define WAVES 4

// ---- workspace layout (floats) ----
// M0eff : [36][32]  (cols 18..31 zero)          @ 0      (1152)
// M1eff : [20][16]  (rows 18..19, cols 8..15 0) @ 1152   (320)
// pwoff : [32]      (ch 18..31 zero)            @ 1472
// dpw   : [32]                                  @ 1504
// total 1536 floats = 6 KB

__global__ void sephia_precomp(const float* __restrict__ W0, const float* __restrict__ d0,
                               const float* __restrict__ W1, const float* __restrict__ d1,
                               const float* __restrict__ pdbm, float* __restrict__ ws) {
    float* M0    = ws;
    float* M1    = ws + 1152;
    float* pwoff = ws + 1472;
    float* dpw   = ws + 1504;
    const int tid = threadIdx.x;

    // M0eff[i][j] = P_UNITY * d0[t,k] * (W0[t,w,2k] - W0[t,w,2k+1]) on the block diagonal
    for (int idx = tid; idx < 36 * 32; idx += blockDim.x) {
        int i = idx >> 5, j = idx & 31;
        float v = 0.0f;
        if (j < 18 && (i / 18) == (j / 9)) {
            int t = j / 9, k = j % 9, w = i % 18;
            float wp = W0[(t * 18 + w) * 18 + 2 * k];
            float wm = W0[(t * 18 + w) * 18 + 2 * k + 1];
            v = 1e-4f * d0[t * 9 + k] * (wp - wm);
        }
        M0[idx] = v;
    }
    // M1eff[w][k] = d1[k] * (W1[0,w,2k] - W1[0,w,2k+1])
    for (int idx = tid; idx < 20 * 16; idx += blockDim.x) {
        int w = idx >> 4, k = idx & 15;
        float v = 0.0f;
        if (w < 18 && k < 8) {
            float wp = W1[w * 16 + 2 * k];
            float wm = W1[w * 16 + 2 * k + 1];
            v = d1[k] * (wp - wm);
        }
        M1[idx] = v;
    }
    // SEPhIA per-channel constants: pw = pwoff + spk * dpw
    for (int c = tid; c < 32; c += blockDim.x) {
        float po = 0.0f, dp = 0.0f;
        if (c < 18) {
            float amp2  = __powf(10.0f, pdbm[c] * 0.1f) * 1000.0f; // dbm_to_watt * 1e6
            float gamma = 0.1f;                                    // 10^(-20/20)
            float wl    = 1550.0f + 0.8f * (float)c;
            float fwhm  = wl * 1e3f / 15000.0f;
            float delta = -250.0f / (0.5f * fwhm);
            float on    = amp2 * (delta * delta + gamma * gamma) / (delta * delta + 1.0f);
            po = amp2 * (gamma * gamma);
            dp = on - po;
        }
        pwoff[c] = po;
        dpw[c]   = dp;
    }
}

__global__ __launch_bounds__(WAVES * 32) void sephia_snn(const float* __restrict__ x,
                                                         const float* __restrict__ ws,
                                                         float* __restrict__ out) {
    __shared__ float lds[WAVES][16 * 20];

    const float* M0    = ws;
    const float* M1    = ws + 1152;
    const float* pwoff = ws + 1472;
    const float* dpw   = ws + 1504;

    const int tid  = threadIdx.x;
    const int lane = tid & 31;
    const int wid  = tid >> 5;
    const int l16  = lane & 15;
    const int lhi  = lane >> 4;              // 0 or 1
    const int rowbase = (blockIdx.x * WAVES + wid) * 16;
    float* sm = &lds[wid][0];

    // zero LDS pad columns 18,19 once (K-padding for layer-1 A tile)
    if (lane < 16) { sm[lane * 20 + 18] = 0.0f; sm[lane * 20 + 19] = 0.0f; }

    // ---- load B matrices into WMMA B-layout registers ----
    // B layout (K x 16, f32): .x = row K = 4*kk + 2*lhi, .y = row +1, col = l16
    v2f B0[9][2];
#pragma unroll
    for (int kk = 0; kk < 9; ++kk) {
        int r = 4 * kk + 2 * lhi;
#pragma unroll
        for (int n = 0; n < 2; ++n) {
            int c = n * 16 + l16;
            B0[kk][n].x = M0[r * 32 + c];
            B0[kk][n].y = M0[(r + 1) * 32 + c];
        }
    }
    v2f B1[5];
#pragma unroll
    for (int kk = 0; kk < 5; ++kk) {
        int r = 4 * kk + 2 * lhi;
        B1[kk].x = M1[r * 16 + l16];
        B1[kk].y = M1[(r + 1) * 16 + l16];
    }

    // per-lane SEPhIA constants in C-tile layout (tile0: ch=l16, tile1: ch=16+l16)
    const float pw_off0 = pwoff[l16];
    const float dpw0    = dpw[l16];
    const int   ch1     = 16 + l16;
    const float pw_off1 = (ch1 < 18) ? pwoff[ch1] : 0.0f;
    const float dpw1    = (ch1 < 18) ? dpw[ch1] : 0.0f;

    const v8f z = {0.f, 0.f, 0.f, 0.f, 0.f, 0.f, 0.f, 0.f};
    v8f m0a = z, m0b = z, m1 = z;   // persistent membrane state (C layout)

    const size_t OFF_S0 = 0;
    const size_t OFF_S1 = (size_t)NT * NB * 18;
    const size_t OFF_M0 = OFF_S1 + (size_t)NT * NB * 8;
    const size_t OFF_M1 = OFF_M0 + (size_t)NT * NB * 18;

    for (int t = 0; t < NT; ++t) {
        // ---- layer 0: [16x36] @ [36x32] via 9 K-steps x 2 N-tiles ----
        const float* xrow = x + ((size_t)t * NB + (size_t)(rowbase + l16)) * 36 + 2 * lhi;
        v8f c0 = z, c1 = z;
#pragma unroll
        for (int kk = 0; kk < 9; ++kk) {
            v2f a = *reinterpret_cast<const v2f*>(xrow + 4 * kk);  // A: row=l16, K=4kk+2lhi{+1}
            c0 = __builtin_amdgcn_wmma_f32_16x16x4_f32(false, a, false, B0[kk][0],
                                                       (short)0, c0, false, false);
            c1 = __builtin_amdgcn_wmma_f32_16x16x4_f32(false, a, false, B0[kk][1],
                                                       (short)0, c1, false, false);
        }

        // ---- LIF0 + SEPhIA all-pass, write pw0 to LDS (A-staging) + outputs ----
        float* s0spk = out + OFF_S0 + ((size_t)t * NB + rowbase) * 18;
        float* s0mem = out + OFF_M0 + ((size_t)t * NB + rowbase) * 18;
#pragma unroll
        for (int i = 0; i < 8; ++i) {
            const int rl = i + 8 * lhi;      // local batch row of this C element
            {   // channels 0..15
                float mm  = m0a[i];
                float rst = (mm > 0.55f) ? 1.0f : 0.0f;
                float mn  = (0.95f * mm + c0[i]) * (1.0f - rst);
                m0a[i] = mn;
                float spk = (mn > 0.55f) ? 1.0f : 0.0f;
                float pw  = pw_off0 + spk * dpw0;
                sm[rl * 20 + l16]      = pw;
                s0spk[rl * 18 + l16]   = pw;
                s0mem[rl * 18 + l16]   = mn;
            }
            {   // channels 16..17 (cols 18..31 of M0eff are zero -> harmless elsewhere)
                float mm  = m0b[i];
                float rst = (mm > 0.55f) ? 1.0f : 0.0f;
                float mn  = (0.95f * mm + c1[i]) * (1.0f - rst);
                m0b[i] = mn;
                float spk = (mn > 0.55f) ? 1.0f : 0.0f;
                float pw  = pw_off1 + spk * dpw1;
                if (ch1 < 18) {
                    sm[rl * 20 + ch1]    = pw;
                    s0spk[rl * 18 + ch1] = pw;
                    s0mem[rl * 18 + ch1] = mn;
                }
            }
        }

        // ---- layer 1: pw0 [16x20(pad)] @ M1eff [20x16], 5 K-steps ----
        // DS ops are in-order within a wave; same-wave RAW through LDS is safe.
        v8f cc = z;
#pragma unroll
        for (int kk = 0; kk < 5; ++kk) {
            int K = 4 * kk + 2 * lhi;
            v2f a = *reinterpret_cast<const v2f*>(&sm[l16 * 20 + K]);
            cc = __builtin_amdgcn_wmma_f32_16x16x4_f32(false, a, false, B1[kk],
                                                       (short)0, cc, false, false);
        }

        // ---- LIF1 + outputs (cols 0..7 valid) ----
        float* s1spk = out + OFF_S1 + ((size_t)t * NB + rowbase) * 8;
        float* s1mem = out + OFF_M1 + ((size_t)t * NB + rowbase) * 8;
#pragma unroll
        for (int i = 0; i < 8; ++i) {
            float mm  = m1[i];
            float rst = (mm > 0.25f) ? 1.0f : 0.0f;
            float mn  = (0.95f * mm + cc[i]) * (1.0f - rst);
            m1[i] = mn;
            float spk = (mn > 0.25f) ? 1.0f : 0.0f;
            if (l16 < 8) {
                const int rl = i + 8 * lhi;
                s1spk[rl * 8 + l16] = spk;
                s1mem[rl * 8 + l16] = mn;
            }
        }
    }
}

extern "C" void kernel_launch(void* const* d_in, const int* in_sizes, int n_in,
                              void* d_out, int out_size, void* d_ws, size_t ws_size,
                              hipStream_t stream) {
    const float* x    = (const float*)d_in[0];  // [32, 8192, 36]
    const float* W0   = (const float*)d_in[1];  // [2, 18, 18]
    const float* d0   = (const float*)d_in[2];  // [2, 9]
    const float* W1   = (const float*)d_in[3];  // [1, 18, 16]
    const float* d1   = (const float*)d_in[4];  // [1, 8]
    const float* pdbm = (const float*)d_in[5];  // [36]
    float* ws = (float*)d_ws;

    sephia_precomp<<<1, 256, 0, stream>>>(W0, d0, W1, d1, pdbm, ws);
    sephia_snn<<<NB / (16 * WAVES), WAVES * 32, 0, stream>>>(x, ws, (float*)d_out);
}